// LocalSparseAttention_36206574306107
// MI455X (gfx1250) — compile-verified
//
#include <hip/hip_runtime.h>
#include <hip/hip_bf16.h>
#include <math.h>

typedef __attribute__((ext_vector_type(2))) float v2f;
typedef __attribute__((ext_vector_type(4))) float v4f;
typedef __attribute__((ext_vector_type(8))) float v8f;

#define S_LEN   4096
#define BATCH   4
#define DMODEL  128
#define HALF    32
#define QTILE   16
#define NKT     5        // 5 aligned 16-wide key tiles cover [i0-32, i0+48)
#define PSTRIDE 84       // LDS row stride in floats (16x80 P tile, padded)

// ---------------------------------------------------------------------------
// Fill kernel, b128-store version: mask = (|i-j|<=32) ? 1 : 0 ; zero ALL of
// attn (band is overwritten later by the attention kernel, stream-ordered).
// All stores are v4f (global_store_b128) and non-temporal: this ~320 MB of
// write-once data exceeds the 192 MB L2, so keep it from rinsing the cache.
// ---------------------------------------------------------------------------
__global__ void lsa_fill_kernel(v4f* __restrict__ attn4, v4f* __restrict__ mask4)
{
    const size_t total4 = (size_t)S_LEN * S_LEN / 4;      // 4,194,304 float4s
    for (size_t idx = blockIdx.x * (size_t)blockDim.x + threadIdx.x;
         idx < total4;
         idx += (size_t)gridDim.x * blockDim.x) {
        const int i  = (int)(idx >> 10);                  // row (S/4 = 1024 f4/row)
        const int j0 = (int)(idx & 1023) * 4;             // first column of the 4
        v4f m;
        {
            const int d0 = i - (j0 + 0);
            const int d1 = i - (j0 + 1);
            const int d2 = i - (j0 + 2);
            const int d3 = i - (j0 + 3);
            m.x = (d0 >= -HALF && d0 <= HALF) ? 1.0f : 0.0f;
            m.y = (d1 >= -HALF && d1 <= HALF) ? 1.0f : 0.0f;
            m.z = (d2 >= -HALF && d2 <= HALF) ? 1.0f : 0.0f;
            m.w = (d3 >= -HALF && d3 <= HALF) ? 1.0f : 0.0f;
        }
        __builtin_nontemporal_store(m, &mask4[idx]);
        const v4f z = {0.f, 0.f, 0.f, 0.f};
        #pragma unroll
        for (int b = 0; b < BATCH; ++b)
            __builtin_nontemporal_store(z, &attn4[(size_t)b * total4 + idx]);
    }
}

// ---------------------------------------------------------------------------
// Banded attention via V_WMMA_F32_16X16X4_F32. One wave32 per 16-query tile.
// ---------------------------------------------------------------------------
__global__ __launch_bounds__(128)
void lsa_attn_kernel(const float* __restrict__ q,
                     const float* __restrict__ k,
                     const float* __restrict__ v,
                     float* __restrict__ out,    // [B,S,D]
                     float* __restrict__ attn)   // [B,S,S]
{
    __shared__ float pbuf[4][QTILE * PSTRIDE];

    const int lane = threadIdx.x & 31;
    const int wave = threadIdx.x >> 5;
    const int tile = blockIdx.x * 4 + wave;          // 0 .. B*S/16 - 1
    const int b    = tile / (S_LEN / QTILE);
    const int i0   = (tile % (S_LEN / QTILE)) * QTILE;

    const int n   = lane & 15;                       // N index (B/C/D lane -> N)
    const int L   = lane >> 4;                       // lane half
    const int off = 2 * L;                           // K-pair offset for A/B frags

    const float scale = 0.08838834764831845f;        // 1/sqrt(128)

    const float* qb = q + (size_t)b * S_LEN * DMODEL;
    const float* kb = k + (size_t)b * S_LEN * DMODEL;
    const float* vb = v + (size_t)b * S_LEN * DMODEL;

    // ---- Q tile as 32 pre-scaled A-fragments (16x4 f32: lane=M, v[]=K pair) ----
    v2f qa[DMODEL / 4];
    {
        const float* qrow = qb + (size_t)(i0 + n) * DMODEL;
        #pragma unroll
        for (int kk = 0; kk < DMODEL / 4; ++kk) {
            float2 t = *(const float2*)(qrow + kk * 4 + off);
            qa[kk].x = t.x * scale;
            qa[kk].y = t.y * scale;
        }
    }

    // ---- scores: 5 tiles of 16x16 = (Q*scale) x K^T, K-dim = 128 ----
    const int j0 = i0 - HALF;                        // 16-aligned window start
    v8f st[NKT];
    #pragma unroll
    for (int t = 0; t < NKT; ++t) {
        const int jr  = j0 + t * 16 + n;
        const int jrc = min(max(jr, 0), S_LEN - 1);  // clamp; masked below
        const float* krow = kb + (size_t)jrc * DMODEL;
        __builtin_prefetch(krow + DMODEL, 0, 1);     // hint next row into GL2
        v8f acc = {};
        #pragma unroll
        for (int kk = 0; kk < DMODEL / 4; ++kk) {
            float2 t2 = *(const float2*)(krow + kk * 4 + off);
            v2f bf; bf.x = t2.x; bf.y = t2.y;        // B 4x16: lane=N, v[]=K pair
            acc = __builtin_amdgcn_wmma_f32_16x16x4_f32(
                false, qa[kk], false, bf, (short)0, acc, false, false);
        }
        st[t] = acc;
    }

    // ---- mask + softmax across the 80-wide window (row in one VGPR slot) ----
    #pragma unroll
    for (int r = 0; r < 8; ++r) {
        const int i = i0 + r + 8 * L;                // C/D: lane-half adds 8 to M
        float mx = -INFINITY;
        #pragma unroll
        for (int t = 0; t < NKT; ++t) {
            const int j = j0 + t * 16 + n;
            const int d = i - j;
            const bool valid = (j >= 0) && (j < S_LEN) && (d >= -HALF) && (d <= HALF);
            const float s = valid ? st[t][r] : -INFINITY;
            st[t][r] = s;
            mx = fmaxf(mx, s);
        }
        #pragma unroll
        for (int x = 1; x < 16; x <<= 1)             // stays inside 16-lane half
            mx = fmaxf(mx, __shfl_xor(mx, x, 32));
        float sum = 0.0f;
        #pragma unroll
        for (int t = 0; t < NKT; ++t) {
            const float p = __expf(st[t][r] - mx);   // exp(-inf)=0 for masked
            st[t][r] = p;
            sum += p;
        }
        #pragma unroll
        for (int x = 1; x < 16; x <<= 1)
            sum += __shfl_xor(sum, x, 32);
        const float inv = 1.0f / sum;
        #pragma unroll
        for (int t = 0; t < NKT; ++t)
            st[t][r] *= inv;
    }

    // ---- write attn band; stage P (16x80) in LDS for the PV matmul ----
    float* pl = pbuf[wave];
    float* attnb = attn + (size_t)b * S_LEN * S_LEN;
    #pragma unroll
    for (int r = 0; r < 8; ++r) {
        const int m = r + 8 * L;
        const int i = i0 + m;
        #pragma unroll
        for (int t = 0; t < NKT; ++t) {
            const int j = j0 + t * 16 + n;
            const int d = i - j;
            const float w = st[t][r];
            if ((j >= 0) && (j < S_LEN) && (d >= -HALF) && (d <= HALF))
                attnb[(size_t)i * S_LEN + j] = w;
            pl[m * PSTRIDE + t * 16 + n] = w;        // masked entries are 0
        }
    }
    __syncthreads();                                 // uniform across all 4 waves

    // ---- output: P (16x80) x V (80x128), 8 d-tiles x 20 k-steps of WMMA ----
    float* ob = out + (size_t)b * S_LEN * DMODEL;
    #pragma unroll
    for (int dt = 0; dt < DMODEL / 16; ++dt) {
        const int d0 = dt * 16;
        v8f acc = {};
        #pragma unroll
        for (int kk = 0; kk < 20; ++kk) {
            const int c = kk * 4 + off;
            v2f af;                                  // A: lane=M (=n), cols c,c+1
            af.x = pl[n * PSTRIDE + c];
            af.y = pl[n * PSTRIDE + c + 1];
            const int jr0  = j0 + c;                 // B: rows c,c+1 of window V
            const int jr0c = min(max(jr0,     0), S_LEN - 1);
            const int jr1c = min(max(jr0 + 1, 0), S_LEN - 1);
            v2f bf;
            bf.x = vb[(size_t)jr0c * DMODEL + d0 + n];
            bf.y = vb[(size_t)jr1c * DMODEL + d0 + n];
            acc = __builtin_amdgcn_wmma_f32_16x16x4_f32(
                false, af, false, bf, (short)0, acc, false, false);
        }
        #pragma unroll
        for (int r = 0; r < 8; ++r) {
            const int i = i0 + r + 8 * L;
            ob[(size_t)i * DMODEL + d0 + n] = acc[r];
        }
    }
}

// ---------------------------------------------------------------------------
extern "C" void kernel_launch(void* const* d_in, const int* in_sizes, int n_in,
                              void* d_out, int out_size, void* d_ws, size_t ws_size,
                              hipStream_t stream)
{
    (void)in_sizes; (void)n_in; (void)out_size; (void)d_ws; (void)ws_size;

    const float* q = (const float*)d_in[0];
    const float* k = (const float*)d_in[1];
    const float* v = (const float*)d_in[2];

    float* out  = (float*)d_out;                              // [B,S,D]
    float* attn = out  + (size_t)BATCH * S_LEN * DMODEL;      // [B,S,S]
    float* mask = attn + (size_t)BATCH * S_LEN * S_LEN;       // [S,S]

    // 1) stream out mask + attn zeros with NT b128 stores (bandwidth-bound bulk)
    lsa_fill_kernel<<<4096, 256, 0, stream>>>((v4f*)attn, (v4f*)mask);

    // 2) banded WMMA attention: 1024 query tiles, 4 wave32 tiles per block
    //    (overwrites the in-band attn entries; stream order guarantees it)
    lsa_attn_kernel<<<(BATCH * S_LEN / QTILE) / 4, 128, 0, stream>>>(q, k, v, out, attn);
}